// EMP_24395414241357
// MI455X (gfx1250) — compile-verified
//
#include <hip/hip_runtime.h>

// ---------------------------------------------------------------------------
// GNN-collapse kernel for MI455X (gfx1250).
//
// reference:  h = segment_sum(features[src] -> dst); r1 = sum_f h;
//             out = int32( 10 * leaky(16 * leaky(r1)) )
// Since the feature-sum commutes with the edge-sum:
//   rowsum[u] = sum_f features[u][f]          (51.2 MB streaming, WMMA f32)
//   hs[v]    += rowsum[src[e]]                (L2-resident scalar atomics)
//   out[v]    = pointwise chain               (400 KB)
// ---------------------------------------------------------------------------

typedef float v2f __attribute__((ext_vector_type(2)));
typedef float v8f __attribute__((ext_vector_type(8)));

constexpr int   D_FEAT    = 128;
constexpr float NEG_SLOPE = 0.1f;

// --- phase 0: zero the float accumulator (aliases d_out) --------------------
__global__ void EMP_zero_f32(float* __restrict__ p, int n) {
  int i = blockIdx.x * blockDim.x + threadIdx.x;
  if (i < n) p[i] = 0.0f;
}

// --- phase 1: rowsum[u] = sum_f features[u][f] via V_WMMA_F32_16X16X4_F32 ---
// One wave = one 16-node tile. A-matrix (16x4 f32) ISA layout:
//   lanes 0-15 : M=lane,    VGPR0=K0, VGPR1=K1
//   lanes 16-31: M=lane-16, VGPR0=K2, VGPR1=K3
// so lane loads float2 at features[row*128 + (lane>>4)*2 + k].
// B = ones(4x16) -> every column of D equals the K-partial rowsum; 32 WMMAs
// accumulate the full 128-wide sum exactly in fp32.
__global__ __launch_bounds__(256) void EMP_rowsum_wmma(
    const float* __restrict__ feat, float* __restrict__ rowsum, int n_nodes) {
  const int lane = threadIdx.x & 31;
  const int wave = threadIdx.x >> 5;
  const int base = (blockIdx.x * 8 + wave) * 16;
  if (base >= n_nodes) return;            // uniform per wave: EXEC stays all-1s

  const int m    = lane & 15;
  const int half = lane >> 4;             // 0: K=k,k+1   1: K=k+2,k+3
  int row = base + m;
  if (row >= n_nodes) row = n_nodes - 1;  // clamp; duplicates are never stored

  const float* rp = feat + (long long)row * D_FEAT + half * 2;

  v8f acc = {};
  v2f ones;
  ones[0] = 1.0f;
  ones[1] = 1.0f;

#pragma unroll
  for (int k = 0; k < D_FEAT; k += 4) {
    v2f a = *reinterpret_cast<const v2f*>(rp + k);
    acc = __builtin_amdgcn_wmma_f32_16x16x4_f32(
        /*neg_a=*/false, a, /*neg_b=*/false, ones,
        /*c_mod=*/(short)0, acc, /*reuse_a=*/false, /*reuse_b=*/false);
  }

  // C/D layout: lanes 0-15 hold M=0..7 in acc[0..7]; lanes 16-31 hold M=8..15.
  // Writers: lanes 0-7 emit M=0..7, lanes 16-23 emit M=8..15.
  const int s = lane & 7;
  float v = acc[0];
  v = (s == 1) ? acc[1] : v;
  v = (s == 2) ? acc[2] : v;
  v = (s == 3) ? acc[3] : v;
  v = (s == 4) ? acc[4] : v;
  v = (s == 5) ? acc[5] : v;
  v = (s == 6) ? acc[6] : v;
  v = (s == 7) ? acc[7] : v;

  if (((lane >> 3) & 1) == 0) {           // lanes 0-7 and 16-23
    int node = base + (lane >> 4) * 8 + s;
    if (node < n_nodes) rowsum[node] = v;
  }
}

// --- phase 2: hs[dst[e]] += rowsum[src[e]] ---------------------------------
// rowsum (400 KB) and hs (400 KB) are L2-resident on a 192 MB L2, so the
// random gather + f32 atomic both stay on-chip.
__global__ void EMP_scatter_add(const int* __restrict__ esrc,
                                const int* __restrict__ edst,
                                const float* __restrict__ rowsum,
                                float* __restrict__ hs, int n_edges) {
  int e = blockIdx.x * blockDim.x + threadIdx.x;
  if (e < n_edges) {
    atomicAdd(&hs[edst[e]], rowsum[esrc[e]]);
  }
}

// --- phase 3: pointwise chain, in place float -> int32 ---------------------
__global__ void EMP_finalize(float* hs, int* out, int n_nodes) {
  int i = blockIdx.x * blockDim.x + threadIdx.x;
  if (i < n_nodes) {
    float r1 = hs[i];
    float a1 = (r1 >= 0.0f) ? r1 : NEG_SLOPE * r1;
    float r2 = 16.0f * a1;
    float a2 = (r2 >= 0.0f) ? r2 : NEG_SLOPE * r2;
    float r3 = 10.0f * a2;
    out[i] = (int)r3;                     // truncation == astype(int32)
  }
}

extern "C" void kernel_launch(void* const* d_in, const int* in_sizes, int n_in,
                              void* d_out, int out_size, void* d_ws, size_t ws_size,
                              hipStream_t stream) {
  const float* feat = (const float*)d_in[0];
  const int*   esrc = (const int*)d_in[1];   // harness convention: integer -> int32
  const int*   edst = (const int*)d_in[2];

  const int n_nodes = in_sizes[0] / D_FEAT;
  const int n_edges = in_sizes[1];

  float* rowsum = (float*)d_ws;              // n_nodes floats (400 KB)
  float* hacc   = (float*)d_out;             // accumulate as f32 in the output buf
  int*   out    = (int*)d_out;               // rewritten in place by finalize

  EMP_zero_f32<<<(n_nodes + 255) / 256, 256, 0, stream>>>(hacc, n_nodes);

  const int tiles  = (n_nodes + 15) / 16;    // 16 nodes per wave
  const int blocks = (tiles + 7) / 8;        // 8 waves per 256-thread block
  EMP_rowsum_wmma<<<blocks, 256, 0, stream>>>(feat, rowsum, n_nodes);

  EMP_scatter_add<<<(n_edges + 255) / 256, 256, 0, stream>>>(esrc, edst, rowsum,
                                                             hacc, n_edges);

  EMP_finalize<<<(n_nodes + 255) / 256, 256, 0, stream>>>(hacc, out, n_nodes);
}